// GNN_16355235463879
// MI455X (gfx1250) — compile-verified
//
#include <hip/hip_runtime.h>

#define NUM_USERS  100000
#define NUM_VENUES 50000
#define DIM        128
#define VFEAT      256
#define NEDGE      1000000

typedef __attribute__((ext_vector_type(2))) float v2f;
typedef __attribute__((ext_vector_type(8))) float v8f;

__device__ __forceinline__ void atomic_add_f32(float* p, float v) {
    __hip_atomic_fetch_add(p, v, __ATOMIC_RELAXED, __HIP_MEMORY_SCOPE_AGENT);
}

// ---------------------------------------------------------------------------
// Zero a region of floats (grid-stride).
// ---------------------------------------------------------------------------
__global__ __launch_bounds__(256) void zero_f32(float* __restrict__ p, long long n) {
    long long i = (long long)blockIdx.x * blockDim.x + threadIdx.x;
    long long stride = (long long)gridDim.x * blockDim.x;
    for (; i < n; i += stride) p[i] = 0.0f;
}

// ---------------------------------------------------------------------------
// v0 = venue_x [NV,256] @ Wp^T [256->128] + b     (fp32 WMMA 16x16x4)
// One wave = one 16x16 output tile. 8 waves/block cover the 8 N-tiles.
// A layout (16x4 f32): lane%16 = row M; lane/16 selects K pair {0,1}/{2,3}.
// C layout: VGPR r -> (row = r + 8*(lane/16), col = lane%16).
// ---------------------------------------------------------------------------
__global__ __launch_bounds__(256) void venue_proj_wmma(
    const float* __restrict__ X,     // [NV, VFEAT]
    const float* __restrict__ W,     // [DIM, VFEAT]  (row n holds weights for out col n)
    const float* __restrict__ bias,  // [DIM]
    float* __restrict__ out)         // [NV, DIM]
{
    const int lane  = threadIdx.x & 31;
    const int wave  = threadIdx.x >> 5;   // N tile 0..7
    const int lm    = lane & 15;
    const int ld    = lane >> 4;          // 0 or 1 (K-pair select)
    const int mBase = blockIdx.x * 16;
    const int nBase = wave * 16;
    const int m = mBase + lm;             // this lane's A row
    const int n = nBase + lm;             // this lane's B column

    const float* __restrict__ arow = X + (size_t)m * VFEAT;
    const float* __restrict__ brow = W + (size_t)n * VFEAT;  // B[k][n] = W[n][k]

    v8f c = {};
    #pragma unroll 8
    for (int k = 0; k < VFEAT; k += 4) {
        const int kk = k + 2 * ld;
        v2f a, b;
        a[0] = arow[kk]; a[1] = arow[kk + 1];
        b[0] = brow[kk]; b[1] = brow[kk + 1];
        c = __builtin_amdgcn_wmma_f32_16x16x4_f32(false, a, false, b,
                                                  (short)0, c, false, false);
    }

    const float bn = bias[n];
    #pragma unroll
    for (int r = 0; r < 8; ++r) {
        const int row = mBase + r + 8 * ld;
        out[(size_t)row * DIM + n] = c[r] + bn;
    }
}

// ---------------------------------------------------------------------------
// Per-edge scatter: agg_v[ev] += u[eu];  agg_u[eu] += v[ev];  counts += 1.
// One wave per edge, each lane owns 4 contiguous dims (float4 gather).
// Tables + accumulators are L2-resident (192 MB) -> atomics stay on-chip.
// ---------------------------------------------------------------------------
__global__ __launch_bounds__(256) void edge_scatter(
    const float* __restrict__ u,      // [NU, DIM]
    const float* __restrict__ v,      // [NV, DIM]
    const int*   __restrict__ eu,     // [E]
    const int*   __restrict__ ev,     // [E]
    float* __restrict__ agg_v,        // [NV, DIM]
    float* __restrict__ agg_u,        // [NU, DIM]
    float* __restrict__ cnt_v,        // [NV]
    float* __restrict__ cnt_u)        // [NU]
{
    const int lane = threadIdx.x & 31;
    const int wave = threadIdx.x >> 5;
    const long long e = (long long)blockIdx.x * 8 + wave;
    if (e >= NEDGE) return;

    const int iu = eu[e];
    const int iv = ev[e];
    const int d  = lane * 4;          // 32 lanes * 4 = 128 dims

    const float4 um = *(const float4*)(u + (size_t)iu * DIM + d);
    const float4 vm = *(const float4*)(v + (size_t)iv * DIM + d);

    float* __restrict__ av = agg_v + (size_t)iv * DIM + d;
    float* __restrict__ au = agg_u + (size_t)iu * DIM + d;
    atomic_add_f32(av + 0, um.x); atomic_add_f32(av + 1, um.y);
    atomic_add_f32(av + 2, um.z); atomic_add_f32(av + 3, um.w);
    atomic_add_f32(au + 0, vm.x); atomic_add_f32(au + 1, vm.y);
    atomic_add_f32(au + 2, vm.z); atomic_add_f32(au + 3, vm.w);

    if (lane == 0) atomic_add_f32(&cnt_v[iv], 1.0f);
    if (lane == 1) atomic_add_f32(&cnt_u[iu], 1.0f);
}

// ---------------------------------------------------------------------------
// out = relu( (agg/max(cnt,1)) @ Wl^T + bl + root @ Wr^T )
// Dual-accumulate fp32 WMMA; mean-normalization folded into the A load
// (each lane's A row is fixed -> one reciprocal per tile per lane).
// ---------------------------------------------------------------------------
__global__ __launch_bounds__(256) void sage_layer_wmma(
    const float* __restrict__ agg,    // [M, DIM] unnormalized sums
    const float* __restrict__ cnt,    // [M]
    const float* __restrict__ root,   // [M, DIM]
    const float* __restrict__ Wl,     // [DIM, DIM]
    const float* __restrict__ bl,     // [DIM]
    const float* __restrict__ Wr,     // [DIM, DIM]
    float* __restrict__ out)          // [M, DIM]
{
    const int lane  = threadIdx.x & 31;
    const int wave  = threadIdx.x >> 5;   // N tile 0..7
    const int lm    = lane & 15;
    const int ld    = lane >> 4;
    const int mBase = blockIdx.x * 16;
    const int nBase = wave * 16;
    const int m = mBase + lm;
    const int n = nBase + lm;

    const float inv = 1.0f / fmaxf(cnt[m], 1.0f);

    const float* __restrict__ a1 = agg  + (size_t)m * DIM;
    const float* __restrict__ a2 = root + (size_t)m * DIM;
    const float* __restrict__ b1 = Wl   + (size_t)n * DIM;   // B[k][n] = Wl[n][k]
    const float* __restrict__ b2 = Wr   + (size_t)n * DIM;

    v8f c = {};
    #pragma unroll 8
    for (int k = 0; k < DIM; k += 4) {
        const int kk = k + 2 * ld;
        v2f A, B;
        A[0] = a1[kk] * inv; A[1] = a1[kk + 1] * inv;
        B[0] = b1[kk];       B[1] = b1[kk + 1];
        c = __builtin_amdgcn_wmma_f32_16x16x4_f32(false, A, false, B,
                                                  (short)0, c, false, false);
        A[0] = a2[kk];       A[1] = a2[kk + 1];
        B[0] = b2[kk];       B[1] = b2[kk + 1];
        c = __builtin_amdgcn_wmma_f32_16x16x4_f32(false, A, false, B,
                                                  (short)0, c, false, false);
    }

    const float bn = bl[n];
    #pragma unroll
    for (int r = 0; r < 8; ++r) {
        const int row = mBase + r + 8 * ld;
        out[(size_t)row * DIM + n] = fmaxf(c[r] + bn, 0.0f);
    }
}

// ---------------------------------------------------------------------------
extern "C" void kernel_launch(void* const* d_in, const int* in_sizes, int n_in,
                              void* d_out, int out_size, void* d_ws, size_t ws_size,
                              hipStream_t stream)
{
    (void)in_sizes; (void)n_in; (void)out_size; (void)ws_size;

    const float* venue_x    = (const float*)d_in[0];
    const float* user_emb   = (const float*)d_in[1];
    const float* proj_w     = (const float*)d_in[2];
    const float* proj_b     = (const float*)d_in[3];
    const float* uv_wl      = (const float*)d_in[4];   // [2,128,128]
    const float* uv_bl      = (const float*)d_in[5];   // [2,128]
    const float* uv_wr      = (const float*)d_in[6];
    const float* vu_wl      = (const float*)d_in[7];
    const float* vu_bl      = (const float*)d_in[8];
    const float* vu_wr      = (const float*)d_in[9];
    const int*   edge_user  = (const int*)d_in[10];
    const int*   edge_venue = (const int*)d_in[11];

    float* out_u = (float*)d_out;                                  // [NU, DIM]
    float* out_v = (float*)d_out + (size_t)NUM_USERS * DIM;        // [NV, DIM]

    // Workspace layout (agg/cnt region kept contiguous for one zero pass)
    float* ws    = (float*)d_ws;
    float* v0    = ws;  ws += (size_t)NUM_VENUES * DIM;
    float* u1    = ws;  ws += (size_t)NUM_USERS  * DIM;
    float* v1    = ws;  ws += (size_t)NUM_VENUES * DIM;
    float* agg_u = ws;  ws += (size_t)NUM_USERS  * DIM;
    float* agg_v = ws;  ws += (size_t)NUM_VENUES * DIM;
    float* cnt_u = ws;  ws += (size_t)NUM_USERS;
    float* cnt_v = ws;  ws += (size_t)NUM_VENUES;
    const long long zlen = (long long)NUM_USERS * DIM + (long long)NUM_VENUES * DIM
                         + NUM_USERS + NUM_VENUES;

    const int W_OFF = DIM * DIM;   // per-layer weight stride
    const int B_OFF = DIM;

    // 0) venue feature projection
    venue_proj_wmma<<<NUM_VENUES / 16, 256, 0, stream>>>(venue_x, proj_w, proj_b, v0);

    // ---- layer 0 ----
    zero_f32<<<2048, 256, 0, stream>>>(agg_u, zlen);
    edge_scatter<<<NEDGE / 8, 256, 0, stream>>>(user_emb, v0, edge_user, edge_venue,
                                                agg_v, agg_u, cnt_v, cnt_u);
    sage_layer_wmma<<<NUM_VENUES / 16, 256, 0, stream>>>(agg_v, cnt_v, v0,
                                                         uv_wl, uv_bl, uv_wr, v1);
    sage_layer_wmma<<<NUM_USERS / 16, 256, 0, stream>>>(agg_u, cnt_u, user_emb,
                                                        vu_wl, vu_bl, vu_wr, u1);

    // ---- layer 1 (writes straight to d_out) ----
    zero_f32<<<2048, 256, 0, stream>>>(agg_u, zlen);
    edge_scatter<<<NEDGE / 8, 256, 0, stream>>>(u1, v1, edge_user, edge_venue,
                                                agg_v, agg_u, cnt_v, cnt_u);
    sage_layer_wmma<<<NUM_VENUES / 16, 256, 0, stream>>>(agg_v, cnt_v, v1,
                                                         uv_wl + W_OFF, uv_bl + B_OFF,
                                                         uv_wr + W_OFF, out_v);
    sage_layer_wmma<<<NUM_USERS / 16, 256, 0, stream>>>(agg_u, cnt_u, u1,
                                                        vu_wl + W_OFF, vu_bl + B_OFF,
                                                        vu_wr + W_OFF, out_u);
}